// ELLGATBlock_51797305589898
// MI455X (gfx1250) — compile-verified
//
#include <hip/hip_runtime.h>
#include <hip/hip_bf16.h>

typedef __attribute__((ext_vector_type(2))) float v2f;
typedef __attribute__((ext_vector_type(8))) float v8f;

#define H_HEADS 4
#define O_FEAT 16
#define K_NBR 32
#define LRELU_SLOPE 0.01f

// ---------------------------------------------------------------------------
// Dual projection GEMM: Qp = Wq[64xKD] * B[KDxN], Kp = Wk[64xKD] * B[KDxN]
// One 16x16 output tile per wave; 4 waves/block cover the 64 output rows for
// one 16-column node tile. B tile staged through LDS, shared by all 4 waves.
// fp32 WMMA (V_WMMA_F32_16X16X4_F32) keeps exact reference precision.
// ---------------------------------------------------------------------------
template <int KD>
__global__ __launch_bounds__(128)
void gemm_dual_kernel(const float* __restrict__ Wq,
                      const float* __restrict__ Wk,
                      const float* __restrict__ B,
                      float* __restrict__ Qp,
                      float* __restrict__ Kp,
                      int N) {
    __shared__ v2f Bt[(KD / 2) * 16];   // [k-pair][n] : (B[2k,n], B[2k+1,n])

    const int tid = threadIdx.x;            // 0..127
    const int ntile = blockIdx.x;
    const int colBase = ntile * 16;

    // Cooperative load of the KDx16 B tile into LDS as float2 k-pairs.
    for (int j = tid; j < (KD / 2) * 16; j += 128) {
        const int kk = j >> 4;              // k-pair index
        const int nn = j & 15;
        int col = colBase + nn;
        if (col >= N) col = N - 1;          // clamp (tiles are exact for N=50000)
        v2f t;
        t.x = B[(2 * kk) * N + col];
        t.y = B[(2 * kk + 1) * N + col];
        Bt[j] = t;
    }
    __syncthreads();

    const int mtile = tid >> 5;             // wave id -> which 16 output rows
    const int lane  = tid & 31;
    const int half  = lane >> 4;            // 0: K=k0,k0+1  1: K=k0+2,k0+3
    const int ml    = lane & 15;            // M (for A) / N (for B,D)

    const float* wq = Wq + (mtile * 16 + ml) * KD;
    const float* wk = Wk + (mtile * 16 + ml) * KD;

    v8f accq = {};
    v8f acck = {};

#pragma unroll
    for (int k0 = 0; k0 < KD; k0 += 4) {
        const int ka = k0 + half * 2;       // even -> 8B aligned
        v2f aq = *(const v2f*)(wq + ka);
        v2f ak = *(const v2f*)(wk + ka);
        v2f b  = Bt[(ka >> 1) * 16 + ml];
        // 8 args: (neg_a, A, neg_b, B, c_mod, C, reuse_a, reuse_b)
        accq = __builtin_amdgcn_wmma_f32_16x16x4_f32(
            false, aq, false, b, (short)0, accq, false, false);
        acck = __builtin_amdgcn_wmma_f32_16x16x4_f32(
            false, ak, false, b, (short)0, acck, false, false);
    }

    const int col = colBase + ml;
    if (col < N) {
#pragma unroll
        for (int v = 0; v < 8; ++v) {
            const int m = mtile * 16 + half * 8 + v;   // D layout: VGPR v -> M=v (+8 hi half)
            Qp[(size_t)m * N + col] = accq[v];
            Kp[(size_t)m * N + col] = acck[v];
        }
    }
}

// ---------------------------------------------------------------------------
// wave32 cross-lane reductions (warpSize == 32 on gfx1250)
// ---------------------------------------------------------------------------
__device__ __forceinline__ float wave_max32(float v) {
#pragma unroll
    for (int off = 16; off > 0; off >>= 1)
        v = fmaxf(v, __shfl_xor(v, off, 32));
    return v;
}
__device__ __forceinline__ float wave_sum32(float v) {
#pragma unroll
    for (int off = 16; off > 0; off >>= 1)
        v += __shfl_xor(v, off, 32);
    return v;
}

// ---------------------------------------------------------------------------
// ELL-GAT attention: one wave per (node, head); lane == neighbor slot.
//   X[o][k]   = leaky_relu(Qp[h,o,n] * Kp[h,o,adj[n,k]])
//   attn[o][] = softmax_k(masked X), NaN->0
//   S[k]      = sum_w attn[w][k] * aw[h][w]
//   out[h,o,n]= sum_k S[k] * Xm[o][k]          (optional fused leaky_relu)
// ---------------------------------------------------------------------------
__global__ __launch_bounds__(256)
void attn_kernel(const int* __restrict__ adj,
                 const float* __restrict__ Qp,
                 const float* __restrict__ Kp,
                 const float* __restrict__ aw,
                 float* __restrict__ Out,
                 int N, int applyLrelu) {
    const int wavesPerBlock = blockDim.x >> 5;
    const int wave = blockIdx.x * wavesPerBlock + (threadIdx.x >> 5);
    const int lane = threadIdx.x & 31;
    if (wave >= N * H_HEADS) return;        // wave-uniform exit

    const int h = wave & (H_HEADS - 1);
    const int n = wave >> 2;                // H_HEADS == 4

    const int nbr = adj[(size_t)n * K_NBR + lane];
    const bool masked = (nbr == -1);
    const int gidx = masked ? 0 : nbr;

    const float* Qbase = Qp + (size_t)(h * O_FEAT) * N;
    const float* Kbase = Kp + (size_t)(h * O_FEAT) * N;

    float X[O_FEAT];
#pragma unroll
    for (int o = 0; o < O_FEAT; ++o) {
        const float q  = Qbase[(size_t)o * N + n];      // broadcast (uniform addr)
        const float kv = Kbase[(size_t)o * N + gidx];   // L2-resident gather
        const float x  = q * kv;
        X[o] = (x > 0.0f) ? x : LRELU_SLOPE * x;
    }

    // S[k] = sum over "w" (== o) of softmax weights times aw
    float S = 0.0f;
#pragma unroll
    for (int o = 0; o < O_FEAT; ++o) {
        const float s = masked ? -3.402823466e38f : X[o];
        const float m = wave_max32(s);
        const float e = masked ? 0.0f : __expf(X[o] - m);
        const float sum = wave_sum32(e);
        const float attn = (sum > 0.0f) ? (e / sum) : 0.0f;   // NaN->0 path
        S += attn * aw[h * O_FEAT + o];
    }

#pragma unroll
    for (int o = 0; o < O_FEAT; ++o) {
        const float contrib = masked ? 0.0f : S * X[o];
        const float r = wave_sum32(contrib);
        if (lane == 0) {
            float v = r;
            if (applyLrelu) v = (v > 0.0f) ? v : LRELU_SLOPE * v;
            Out[(size_t)(h * O_FEAT + o) * N + n] = v;
        }
    }
}

// ---------------------------------------------------------------------------
// Launch. Inputs: adj[N,32] i32, Q[128,N] f32, qw1/kw1[4,16,128], aw1[4,16],
// qw2/kw2[4,16,64], aw2[4,16]. Output: [64, N] f32.
// ---------------------------------------------------------------------------
extern "C" void kernel_launch(void* const* d_in, const int* in_sizes, int n_in,
                              void* d_out, int out_size, void* d_ws, size_t ws_size,
                              hipStream_t stream) {
    const int*   adj = (const int*)d_in[0];
    const float* Q   = (const float*)d_in[1];
    const float* qw1 = (const float*)d_in[2];
    const float* kw1 = (const float*)d_in[3];
    const float* aw1 = (const float*)d_in[4];
    const float* qw2 = (const float*)d_in[5];
    const float* kw2 = (const float*)d_in[6];
    const float* aw2 = (const float*)d_in[7];
    float* out = (float*)d_out;

    const int N = in_sizes[0] / K_NBR;      // 50000

    float* ws  = (float*)d_ws;
    float* Qp1 = ws;
    float* Kp1 = ws + (size_t)64 * N;
    float* Q2  = ws + (size_t)128 * N;      // leaky_relu(out1), layer-2 input
    float* Qp2 = ws + (size_t)192 * N;
    float* Kp2 = ws + (size_t)256 * N;

    const int ntiles = (N + 15) / 16;       // 3125 exact
    const int tasks  = N * H_HEADS;
    const int ablocks = (tasks + 7) / 8;    // 8 waves / 256-thread block

    // Layer 1
    gemm_dual_kernel<128><<<ntiles, 128, 0, stream>>>(qw1, kw1, Q, Qp1, Kp1, N);
    attn_kernel<<<ablocks, 256, 0, stream>>>(adj, Qp1, Kp1, aw1, Q2, N, 1);
    // Layer 2
    gemm_dual_kernel<64><<<ntiles, 128, 0, stream>>>(qw2, kw2, Q2, Qp2, Kp2, N);
    attn_kernel<<<ablocks, 256, 0, stream>>>(adj, Qp2, Kp2, aw2, out, N, 0);
}